// SingleOpAttnBlock_3092376453886
// MI455X (gfx1250) — compile-verified
//
#include <hip/hip_runtime.h>

typedef __attribute__((ext_vector_type(2))) float v2f;
typedef __attribute__((ext_vector_type(8))) float v8f;

#define BATCH      4
#define NROWS      65536
#define NMASK      (NROWS - 1)
#define FDIM       128
#define TILE       64          // output rows per block
#define LDW        136         // W LDS stride (2*136 mod 64 = 16 -> no A/B half-lane bank overlap)
#define LDH        130         // h tile LDS stride (mod 64 = 2 -> conflict-free row-per-lane dots)
#define LDG        132         // g tile LDS stride (mod 64 = 4 -> conflict-free A-fragment reads)
#define LEAKY      0.2f
#define NEG_INF_F  -9e15f

// LDS layout (floats):
//   sW  [128][LDW]  sH [66][LDH]  sG [64][LDG]  wa1[128] wa2[128] s1[66] s2[66] att[64][4]
#define SMEM_FLOATS (FDIM*LDW + (TILE+2)*LDH + TILE*LDG + FDIM + FDIM + (TILE+2) + (TILE+2) + TILE*4)

__global__ __launch_bounds__(256, 2)
void gat_window_wmma_kernel(const float* __restrict__ h,
                            const int*   __restrict__ op_mask,
                            const float* __restrict__ W,
                            const float* __restrict__ a,
                            float*       __restrict__ out)
{
    extern __shared__ float smem[];
    float* sW  = smem;                        // 128*LDW
    float* sH  = sW  + FDIM * LDW;            // 66*LDH
    float* sG  = sH  + (TILE + 2) * LDH;      // 64*LDG
    float* wa1 = sG  + TILE * LDG;            // 128
    float* wa2 = wa1 + FDIM;                  // 128
    float* s1  = wa2 + FDIM;                  // 66
    float* s2  = s1  + (TILE + 2);            // 66
    float* att = s2  + (TILE + 2);            // 64*4

    const int tid   = threadIdx.x;
    const int b     = blockIdx.x >> 10;       // 1024 tiles per batch
    const int tile  = blockIdx.x & 1023;
    const int start = tile * TILE;

    // ---- Phase 0: stage W into LDS (coalesced) ----
    for (int idx = tid; idx < FDIM * FDIM; idx += 256) {
        int r = idx >> 7, c = idx & 127;
        sW[r * LDW + c] = W[idx];
    }
    // ---- Phase 1: stage h rows [start-1, start+TILE] (wrap within batch) ----
    const float* hb = h + (size_t)b * NROWS * FDIM;
    for (int idx = tid; idx < (TILE + 2) * FDIM; idx += 256) {
        int r = idx >> 7, c = idx & 127;
        int gr = (start - 1 + r) & NMASK;
        sH[r * LDH + c] = hb[(size_t)gr * FDIM + c];
    }
    __syncthreads();

    // ---- Phase 2a: wa1 = W @ a[:F], wa2 = W @ a[F:] ----
    if (tid < 2 * FDIM) {
        const float* av = a + ((tid < FDIM) ? 0 : FDIM);
        int r = tid & (FDIM - 1);
        float acc = 0.f;
        #pragma unroll 8
        for (int j = 0; j < FDIM; ++j) acc += sW[r * LDW + j] * av[j];
        if (tid < FDIM) wa1[r] = acc; else wa2[r] = acc;
    }
    __syncthreads();

    // ---- Phase 2b: per-staged-row logits s1 = h.wa1, s2 = h.wa2 ----
    if (tid < TILE + 2) {
        const float* hr = &sH[tid * LDH];
        float acc1 = 0.f, acc2 = 0.f;
        #pragma unroll 8
        for (int k = 0; k < FDIM; ++k) {
            float x = hr[k];
            acc1 += x * wa1[k];
            acc2 += x * wa2[k];
        }
        s1[tid] = acc1;
        s2[tid] = acc2;
    }
    __syncthreads();

    // ---- Phase 3: leaky-relu, mask, 3-way softmax ----
    if (tid < TILE) {
        int   gr = start + tid;
        float c1 = s1[tid + 1];
        float ep = c1 + s2[tid];       // prev  (roll +1)
        float es = c1 + s2[tid + 1];   // self
        float en = c1 + s2[tid + 2];   // next  (roll -1)
        ep = ep > 0.f ? ep : LEAKY * ep;
        es = es > 0.f ? es : LEAKY * es;
        en = en > 0.f ? en : LEAKY * en;
        const int* m = op_mask + ((size_t)(b * NROWS + gr)) * 3;
        if (m[0] > 0) ep = NEG_INF_F;
        if (m[1] > 0) es = NEG_INF_F;
        if (m[2] > 0) en = NEG_INF_F;
        float mx = fmaxf(ep, fmaxf(es, en));
        float w0 = __expf(ep - mx), w1 = __expf(es - mx), w2 = __expf(en - mx);
        float inv = 1.f / (w0 + w1 + w2);
        att[tid * 4 + 0] = w0 * inv;
        att[tid * 4 + 1] = w1 * inv;
        att[tid * 4 + 2] = w2 * inv;
    }
    __syncthreads();

    // ---- Phase 4: blended tile g[r] = a0*h[r-1] + a1*h[r] + a2*h[r+1] ----
    for (int idx = tid; idx < TILE * FDIM; idx += 256) {
        int r = idx >> 7, c = idx & 127;
        float a0 = att[r * 4 + 0], a1v = att[r * 4 + 1], a2v = att[r * 4 + 2];
        sG[r * LDG + c] = a0  * sH[(r    ) * LDH + c]
                        + a1v * sH[(r + 1) * LDH + c]
                        + a2v * sH[(r + 2) * LDH + c];
    }
    __syncthreads();

    // ---- Phase 5: out_tile = g @ W via V_WMMA_F32_16X16X4_F32 ----
    const int wave  = tid >> 5;
    const int lane  = tid & 31;
    const int mtile = wave & 3;            // row-tile 0..3 (16 rows each)
    const int nset  = (wave >> 2) * 4;     // col-tile set {0..3} or {4..7}
    const int lrow  = lane & 15;
    const int khalf = (lane >> 4) << 1;    // 0 (lanes 0-15) or 2 (lanes 16-31)

    // Hoist all 32 A-fragments for this wave's row-tile (16xK A: VGPR0=K0/K2, VGPR1=K1/K3)
    v2f afr[32];
    {
        const float* grow = &sG[(mtile * 16 + lrow) * LDG];
        #pragma unroll
        for (int k = 0; k < 32; ++k) {
            afr[k].x = grow[4 * k + khalf];
            afr[k].y = grow[4 * k + khalf + 1];
        }
    }

    float* outb = out + ((size_t)(b * NROWS + start)) * FDIM;
    for (int nt = nset; nt < nset + 4; ++nt) {
        v8f acc = {};
        const float* wb = &sW[nt * 16 + lrow];
        #pragma unroll
        for (int k = 0; k < 32; ++k) {
            v2f bf;   // 4x16 B: VGPR0 = K=(0|2), VGPR1 = K=(1|3), N = lane&15
            bf.x = wb[(4 * k + khalf)     * LDW];
            bf.y = wb[(4 * k + khalf + 1) * LDW];
            acc = __builtin_amdgcn_wmma_f32_16x16x4_f32(
                      /*neg_a=*/false, afr[k], /*neg_b=*/false, bf,
                      /*c_mod=*/(short)0, acc, /*reuse_a=*/false, /*reuse_b=*/false);
        }
        // 16x16 f32 D layout: VGPR v -> M=v (lanes 0-15) / M=v+8 (lanes 16-31), N=lane&15
        int colg  = nt * 16 + lrow;
        int rbase = mtile * 16 + ((lane >> 4) * 8);
        #pragma unroll
        for (int v = 0; v < 8; ++v) {
            outb[(size_t)(rbase + v) * FDIM + colg] = acc[v];
        }
    }
}

extern "C" void kernel_launch(void* const* d_in, const int* in_sizes, int n_in,
                              void* d_out, int out_size, void* d_ws, size_t ws_size,
                              hipStream_t stream) {
    const float* h    = (const float*)d_in[0];   // [4, 65536, 128] f32
    const int*   mask = (const int*)  d_in[1];   // [4, 65536, 3]   i32
    const float* W    = (const float*)d_in[2];   // [128, 128]      f32
    const float* a    = (const float*)d_in[3];   // [256, 1]        f32
    float*       out  = (float*)d_out;           // [4, 65536, 128] f32

    const size_t smem_bytes = (size_t)SMEM_FLOATS * sizeof(float);  // ~140 KB -> 2 blocks/WGP
    dim3 grid(BATCH * (NROWS / TILE));   // 4096 blocks
    dim3 block(256);                     // 8 waves (wave32)
    gat_window_wmma_kernel<<<grid, block, smem_bytes, stream>>>(h, mask, W, a, out);
}